// ModelNew_25056839205395
// MI455X (gfx1250) — compile-verified
//
#include <hip/hip_runtime.h>
#include <stdint.h>

// N=32, C=32, D=16, H=32, W=64  ->  out (32,32,8,16,32)
// One wave32 per T=4 consecutive pooled output rows, double-buffered
// global->LDS async pipeline (ASYNCcnt).

typedef __attribute__((ext_vector_type(4))) float fv4;
typedef __attribute__((ext_vector_type(2))) float fv2;

#define WPB 8               // waves per block
#define TPB (WPB * 32)      // 256 threads
#define TILES 4             // output rows per wave (share nc, d2; h2 advances)

// Issue one tile: two 512B async copies (rows (d,h..h+1) and (d+1,h..h+1)).
__device__ __forceinline__ void issue_tile(const float* gbase, uint32_t lds_off, int lane)
{
    uint64_t g0 = (uint64_t)(uintptr_t)(gbase + 4 * lane);
    uint64_t g1 = (uint64_t)(uintptr_t)(gbase + 32LL * 64 + 4 * lane); // +1 D plane
    uint32_t l0 = lds_off + 16u * lane;
    uint32_t l1 = l0 + 512u;
    asm volatile("global_load_async_to_lds_b128 %0, %1, off"
                 :: "v"(l0), "v"(g0) : "memory");
    asm volatile("global_load_async_to_lds_b128 %0, %1, off"
                 :: "v"(l1), "v"(g1) : "memory");
}

__device__ __forceinline__ void compute_tile(const float* tbuf, int lane, int m,
                                             float sw, fv4 g, fv4 bt,
                                             float* __restrict__ out, long long wid)
{
    // Lanes 0-15: rows (d,h)/(d+1,h); lanes 16-31: rows (d,h+1)/(d+1,h+1)
    fv4 a = *(const fv4*)(tbuf + 4 * lane);         // chunk0, w = 4m..4m+3
    fv4 b = *(const fv4*)(tbuf + 128 + 4 * lane);   // chunk1, same w range

    float va0 = a.x + sw, va1 = a.y + sw, va2 = a.z + sw, va3 = a.w + sw;
    float vb0 = b.x + sw, vb1 = b.y + sw, vb2 = b.z + sw, vb3 = b.w + sw;

    float sA = va0 + va1 + va2 + va3;
    float qA = va0*va0 + va1*va1 + va2*va2 + va3*va3;
    float sB = vb0 + vb1 + vb2 + vb3;
    float qB = vb0*vb0 + vb1*vb1 + vb2*vb2 + vb3*vb3;

#pragma unroll
    for (int mask = 1; mask <= 8; mask <<= 1) {   // reduce within each 16-lane half
        sA += __shfl_xor(sA, mask, 32);
        qA += __shfl_xor(qA, mask, 32);
        sB += __shfl_xor(sB, mask, 32);
        qB += __shfl_xor(qB, mask, 32);
    }

    const float inv64 = 1.0f / 64.0f;
    float meanA = sA * inv64;
    float rA    = rsqrtf(qA * inv64 - meanA * meanA + 1e-5f);
    float meanB = sB * inv64;
    float rB    = rsqrtf(qB * inv64 - meanB * meanB + 1e-5f);

    float lnA0 = (va0 - meanA) * rA * g.x + bt.x;
    float lnA1 = (va1 - meanA) * rA * g.y + bt.y;
    float lnA2 = (va2 - meanA) * rA * g.z + bt.z;
    float lnA3 = (va3 - meanA) * rA * g.w + bt.w;
    float lnB0 = (vb0 - meanB) * rB * g.x + bt.x;
    float lnB1 = (vb1 - meanB) * rB * g.y + bt.y;
    float lnB2 = (vb2 - meanB) * rB * g.z + bt.z;
    float lnB3 = (vb3 - meanB) * rB * g.w + bt.w;

    float s0 = lnA0 + lnA1 + lnB0 + lnB1;   // pooled w2 = 2m
    float s1 = lnA2 + lnA3 + lnB2 + lnB3;   // pooled w2 = 2m+1
    s0 += __shfl_xor(s0, 16, 32);           // merge the two H rows
    s1 += __shfl_xor(s1, 16, 32);

    float p0 = s0 * 0.125f;
    float p1 = s1 * 0.125f;
    const float ks = 0.7071067811865476f;
    float o0 = 0.5f * p0 * (1.0f + erff(p0 * ks));
    float o1 = 0.5f * p1 * (1.0f + erff(p1 * ks));

    if (lane < 16) {
        fv2 o; o.x = o0; o.y = o1;
        __builtin_nontemporal_store(o, (fv2*)(out + wid * 32 + 2 * m));
    }
}

__global__ __launch_bounds__(TPB)
void ln_pool_gelu_kernel(const float* __restrict__ x,
                         const float* __restrict__ sum_weight,
                         const float* __restrict__ gamma,
                         const float* __restrict__ beta,
                         float* __restrict__ out)
{
    // Per wave: 2 buffers x 256 floats (double buffer). 16 KB per block.
    __shared__ __align__(16) float buf[WPB * 2 * 256];

    const int tid  = threadIdx.x;
    const int lane = tid & 31;
    const int wave = tid >> 5;
    // Base output-row id for this wave's TILES consecutive rows.
    const long long wbase = ((long long)blockIdx.x * WPB + wave) * TILES; // 0..131068

    const int h2b = (int)(wbase & 15);          // h2 for tile 0 (tiles stay within same 16-group)
    const int d2  = (int)((wbase >> 4) & 7);
    const int nc  = (int)(wbase >> 7);          // n*32 + c

    const int d = d2 << 1;
    const int h = h2b << 1;

    // chunk0 base of tile 0; each tile advances by 2 H-rows = 128 floats.
    const float* gbase0 = x + (((long long)nc * 16 + d) * 32 + h) * 64;

    float* wbuf = &buf[wave * 512];
    const uint32_t lds_base = (uint32_t)(uintptr_t)wbuf;  // LDS byte offset (low 32 bits)

    // Hoisted uniforms.
    const float sw = sum_weight[0];
    const int m = lane & 15;
    fv4 g  = *(const fv4*)(gamma + 4 * m);
    fv4 bt = *(const fv4*)(beta  + 4 * m);

    // ---- double-buffered async pipeline ----
    issue_tile(gbase0, lds_base, lane);                    // tile 0 -> buffer 0

#pragma unroll
    for (int k = 0; k < TILES; ++k) {
        if (k + 1 < TILES) {
            issue_tile(gbase0 + (long long)(k + 1) * 128,
                       lds_base + ((k + 1) & 1) * 1024u, lane);
            // In-order ASYNCcnt: <=2 outstanding means tile k's 2 copies landed.
            asm volatile("s_wait_asynccnt 2" ::: "memory");
        } else {
            asm volatile("s_wait_asynccnt 0" ::: "memory");
        }
        compute_tile(wbuf + (k & 1) * 256, lane, m, sw, g, bt, out, wbase + k);
    }
}

extern "C" void kernel_launch(void* const* d_in, const int* in_sizes, int n_in,
                              void* d_out, int out_size, void* d_ws, size_t ws_size,
                              hipStream_t stream) {
    (void)in_sizes; (void)n_in; (void)out_size; (void)d_ws; (void)ws_size;
    const float* x     = (const float*)d_in[0];
    const float* sw    = (const float*)d_in[1];
    const float* gamma = (const float*)d_in[2];
    const float* beta  = (const float*)d_in[3];
    float* out = (float*)d_out;

    const int rows   = 32 * 32 * 8 * 16;        // 131072 pooled output rows
    const int waves  = rows / TILES;            // 32768
    const int blocks = waves / WPB;             // 4096
    hipLaunchKernelGGL(ln_pool_gelu_kernel, dim3(blocks), dim3(TPB), 0, stream,
                       x, sw, gamma, beta, out);
}